// TaskAlignedAssigner_35828617183980
// MI455X (gfx1250) — compile-verified
//
#include <hip/hip_runtime.h>
#include <hip/hip_bf16.h>
#include <math.h>

#define BS   32
#define NA   8400
#define NC   80
#define NB   64
#define NTOP 13

typedef float v2f __attribute__((ext_vector_type(2)));
typedef float v8f __attribute__((ext_vector_type(8)));

// ---------------------------------------------------------------------------
// CIoU (box1 = gt, box2 = pred), clipped at 0, matching the reference math.
// ---------------------------------------------------------------------------
__device__ __forceinline__ float ciou_clip(float gx1, float gy1, float gx2, float gy2,
                                           float px1, float py1, float px2, float py2,
                                           float w2, float h2, float at2)
{
    float w1 = gx2 - gx1, h1 = gy2 - gy1 + 1e-7f;
    float iw = fminf(gx2, px2) - fmaxf(gx1, px1);
    float ih = fminf(gy2, py2) - fmaxf(gy1, py1);
    float inter = fmaxf(iw, 0.0f) * fmaxf(ih, 0.0f);
    float uni   = w1 * h1 + w2 * h2 - inter + 1e-7f;
    float iou   = inter / uni;
    float cw = fmaxf(gx2, px2) - fminf(gx1, px1);
    float ch = fmaxf(gy2, py2) - fminf(gy1, py1);
    float c2 = cw * cw + ch * ch + 1e-7f;
    float dx = px1 + px2 - gx1 - gx2;
    float dy = py1 + py2 - gy1 - gy2;
    float rho2 = (dx * dx + dy * dy) * 0.25f;
    float dv = at2 - atanf(w1 / h1);
    float vv = 0.40528473456935109f * dv * dv;   // 4/pi^2
    float alpha = vv / (vv - iou + 1.0f + 1e-7f);
    float ci = iou - (rho2 / c2 + vv * alpha);
    return fmaxf(ci, 0.0f);
}

// ---------------------------------------------------------------------------
// K0: zero the per-(b,gt) positive-maximum accumulators (2 * BS*NB uints).
// ---------------------------------------------------------------------------
__global__ void k_init(unsigned int* __restrict__ p, int n)
{
    int i = blockIdx.x * blockDim.x + threadIdx.x;
    if (i < n) p[i] = 0u;
}

// ---------------------------------------------------------------------------
// K1: fused score-gather (one-hot GEMM via V_WMMA_F32_16X16X4_F32) + CIoU.
// One wave handles a 16-gt x 16-anchor tile; D-matrix VGPR v holds
// S[gt = gtb + 8*(lane>=16) + v, anchor = ancb + (lane&15)].
// ---------------------------------------------------------------------------
__global__ __launch_bounds__(32) void k_align(
    const float* __restrict__ pd_scores, const float* __restrict__ pd_bboxes,
    const float* __restrict__ anc,       const int*   __restrict__ gt_labels,
    const float* __restrict__ gt_bboxes, const float* __restrict__ mask_gt,
    float* __restrict__ g_align, float* __restrict__ g_ovl)
{
    const int lane = threadIdx.x;
    const int hi   = lane >> 4;     // which K-pair (A) / K-row pair (B)
    const int lo   = lane & 15;     // M for A, N for B/C/D
    const int ancb = blockIdx.x * 16;
    const int gtb  = blockIdx.y * 16;
    const int b    = blockIdx.z;

    const int    a     = ancb + lo;
    const int    mylbl = gt_labels[b * NB + gtb + lo];
    const float* prow  = pd_scores + ((size_t)b * NA + a) * NC;

    v8f c = {0.f, 0.f, 0.f, 0.f, 0.f, 0.f, 0.f, 0.f};

#if __has_builtin(__builtin_amdgcn_wmma_f32_16x16x4_f32)
    // Exact gather-as-GEMM: A = one-hot(labels) 16x4 slice, B = scores 4x16 slice.
    for (int k0 = 0; k0 < NC; k0 += 4) {
        const int ka = k0 + hi * 2;
        v2f av, bv;
        av.x = (mylbl == ka    ) ? 1.0f : 0.0f;
        av.y = (mylbl == ka + 1) ? 1.0f : 0.0f;
        bv.x = prow[ka];
        bv.y = prow[ka + 1];
        c = __builtin_amdgcn_wmma_f32_16x16x4_f32(false, av, false, bv,
                                                  (short)0, c, false, false);
    }
#else
    // Fallback: direct gather in the same D layout.
    for (int v = 0; v < 8; ++v) {
        int gt = gtb + hi * 8 + v;
        int l  = gt_labels[b * NB + gt];
        c[v] = pd_scores[((size_t)b * NA + a) * NC + l];
    }
#endif

    // Per-lane geometry for anchor `a`.
    const float* pb = pd_bboxes + ((size_t)b * NA + a) * 4;
    float px1 = pb[0], py1 = pb[1], px2 = pb[2], py2 = pb[3];
    float ax = anc[a * 2 + 0], ay = anc[a * 2 + 1];
    float w2 = px2 - px1, h2 = py2 - py1 + 1e-7f;
    float at2 = atanf(w2 / h2);

    for (int v = 0; v < 8; ++v) {
        int gt = gtb + hi * 8 + v;
        const float* gb = gt_bboxes + ((size_t)b * NB + gt) * 4;
        float gx1 = gb[0], gy1 = gb[1], gx2 = gb[2], gy2 = gb[3];
        float mg = mask_gt[b * NB + gt];
        float dmin = fminf(fminf(ax - gx1, ay - gy1), fminf(gx2 - ax, gy2 - ay));
        bool valid = (dmin > 1e-9f) && (mg > 0.0f);
        float ov = 0.0f, al = 0.0f;
        if (valid) {
            ov = ciou_clip(gx1, gy1, gx2, gy2, px1, py1, px2, py2, w2, h2, at2);
            float o2 = ov * ov;
            al = c[v] * (o2 * o2 * o2);   // score^1 * overlap^6
        }
        size_t idx = ((size_t)(b * NB + gt)) * NA + a;
        g_align[idx] = al;
        g_ovl[idx]   = ov;
    }
}

// ---------------------------------------------------------------------------
// K2: per-(b,gt) top-13 of the align row; ties -> lowest index (jax top_k).
// ---------------------------------------------------------------------------
__global__ __launch_bounds__(256) void k_topk(
    const float* __restrict__ g_align, const float* __restrict__ mask_gt,
    unsigned char* __restrict__ mtop)
{
    const int gt = blockIdx.x;
    const int b  = blockIdx.y;
    const float* row = g_align + ((size_t)(b * NB + gt)) * NA;
    unsigned char* mrow = mtop + ((size_t)(b * NB + gt)) * NA;
    const int tid = threadIdx.x;

    for (int a = tid; a < NA; a += 256) mrow[a] = 0;
    __syncthreads();
    if (mask_gt[b * NB + gt] <= 0.0f) return;   // uniform per block

    __shared__ float sv[256];
    __shared__ int   si[256];
    __shared__ int   ssel[NTOP];

    for (int it = 0; it < NTOP; ++it) {
        float bv = -1.0f;
        int   bi = 0;
        for (int a = tid; a < NA; a += 256) {
            bool taken = false;
            for (int s = 0; s < it; ++s)
                if (ssel[s] == a) { taken = true; break; }
            if (taken) continue;
            float v = row[a];
            if (v > bv || (v == bv && a < bi)) { bv = v; bi = a; }
        }
        sv[tid] = bv; si[tid] = bi;
        __syncthreads();
        for (int off = 128; off > 0; off >>= 1) {
            if (tid < off) {
                float v2 = sv[tid + off]; int i2 = si[tid + off];
                if (v2 > sv[tid] || (v2 == sv[tid] && i2 < si[tid])) {
                    sv[tid] = v2; si[tid] = i2;
                }
            }
            __syncthreads();
        }
        if (tid == 0) { ssel[it] = si[0]; mrow[si[0]] = 1; }
        __syncthreads();
    }
}

// ---------------------------------------------------------------------------
// K3: per-anchor resolution -> fg, target_gt_idx, labels, bboxes; and
// per-gt positive maxima via uint-punned atomicMax (all values >= 0).
// ---------------------------------------------------------------------------
__global__ __launch_bounds__(256) void k_resolve(
    const float* __restrict__ gt_bboxes, const int* __restrict__ gt_labels,
    const float* __restrict__ mask_gt,   const float* __restrict__ anc,
    const unsigned char* __restrict__ mtop,
    const float* __restrict__ g_align, const float* __restrict__ g_ovl,
    float* __restrict__ out_labels, float* __restrict__ out_bboxes,
    float* __restrict__ out_fg,     float* __restrict__ out_tgt,
    unsigned int* __restrict__ pos_align, unsigned int* __restrict__ pos_over)
{
    const int b = blockIdx.y;
    const int a = blockIdx.x * 256 + threadIdx.x;

    __shared__ float sgt[NB * 4];
    __shared__ float smg[NB];
    __shared__ int   slb[NB];
    for (int i = threadIdx.x; i < NB * 4; i += 256) sgt[i] = gt_bboxes[b * NB * 4 + i];
    for (int i = threadIdx.x; i < NB;     i += 256) {
        smg[i] = mask_gt[b * NB + i];
        slb[i] = gt_labels[b * NB + i];
    }
    __syncthreads();
    if (a >= NA) return;

    float ax = anc[a * 2 + 0], ay = anc[a * 2 + 1];
    int fg = 0, tgt = 0;
    for (int gt = 0; gt < NB; ++gt) {
        if (mtop[((size_t)(b * NB + gt)) * NA + a]) {
            float gx1 = sgt[gt * 4 + 0], gy1 = sgt[gt * 4 + 1];
            float gx2 = sgt[gt * 4 + 2], gy2 = sgt[gt * 4 + 3];
            float dmin = fminf(fminf(ax - gx1, ay - gy1), fminf(gx2 - ax, gy2 - ay));
            if (dmin > 1e-9f && smg[gt] > 0.0f) { fg++; if (fg == 1) tgt = gt; }
        }
    }
    if (fg > 1) {   // multi-assignment: pick gt with max overlap (first max)
        float bv = -1.0f; int bi = 0;
        for (int gt = 0; gt < NB; ++gt) {
            float o = g_ovl[((size_t)(b * NB + gt)) * NA + a];
            if (o > bv) { bv = o; bi = gt; }
        }
        tgt = bi; fg = 1;
    }

    size_t oi = (size_t)b * NA + a;
    out_tgt[oi] = (float)tgt;
    out_fg[oi]  = fg ? 1.0f : 0.0f;
    int lb = slb[tgt]; if (lb < 0) lb = 0;
    out_labels[oi] = (float)lb;
    float4 tb = make_float4(sgt[tgt * 4 + 0], sgt[tgt * 4 + 1],
                            sgt[tgt * 4 + 2], sgt[tgt * 4 + 3]);
    *(float4*)(out_bboxes + oi * 4) = tb;

    if (fg) {
        size_t ci = ((size_t)(b * NB + tgt)) * NA + a;
        atomicMax(&pos_align[b * NB + tgt], __float_as_uint(g_align[ci]));
        atomicMax(&pos_over [b * NB + tgt], __float_as_uint(g_ovl[ci]));
    }
}

// ---------------------------------------------------------------------------
// K4: normalized one-hot target_scores.
// ---------------------------------------------------------------------------
__global__ __launch_bounds__(256) void k_scores(
    const float* __restrict__ g_align,
    const unsigned int* __restrict__ pos_align, const unsigned int* __restrict__ pos_over,
    const float* __restrict__ out_fg, const float* __restrict__ out_tgt,
    const float* __restrict__ out_labels, float* __restrict__ out_scores)
{
    const int b = blockIdx.y;
    const int a = blockIdx.x * 256 + threadIdx.x;
    if (a >= NA) return;
    size_t oi = (size_t)b * NA + a;

    float norm = 0.0f;
    if (out_fg[oi] > 0.0f) {
        int tgt = (int)out_tgt[oi];
        float pa = __uint_as_float(pos_align[b * NB + tgt]);
        float po = __uint_as_float(pos_over [b * NB + tgt]);
        norm = (g_align[((size_t)(b * NB + tgt)) * NA + a] * po) / (pa + 1e-9f);
    }
    int lb = (int)out_labels[oi];
    float* dst = out_scores + oi * NC;
    for (int c4 = 0; c4 < NC; c4 += 4) {
        float4 t;
        t.x = (c4 + 0 == lb) ? norm : 0.0f;
        t.y = (c4 + 1 == lb) ? norm : 0.0f;
        t.z = (c4 + 2 == lb) ? norm : 0.0f;
        t.w = (c4 + 3 == lb) ? norm : 0.0f;
        *(float4*)(dst + c4) = t;
    }
}

// ---------------------------------------------------------------------------
// Host launcher
// ---------------------------------------------------------------------------
extern "C" void kernel_launch(void* const* d_in, const int* in_sizes, int n_in,
                              void* d_out, int out_size, void* d_ws, size_t ws_size,
                              hipStream_t stream)
{
    (void)in_sizes; (void)n_in; (void)out_size; (void)ws_size;

    const float* pd_scores = (const float*)d_in[0];
    const float* pd_bboxes = (const float*)d_in[1];
    const float* anc       = (const float*)d_in[2];
    const int*   gt_labels = (const int*)  d_in[3];
    const float* gt_bboxes = (const float*)d_in[4];
    const float* mask_gt   = (const float*)d_in[5];

    // Output layout (floats), tuple order:
    // labels | bboxes | scores | fg_mask | target_gt_idx
    float* out = (float*)d_out;
    float* out_labels = out;
    float* out_bboxes = out_labels + (size_t)BS * NA;
    float* out_scores = out_bboxes + (size_t)BS * NA * 4;
    float* out_fg     = out_scores + (size_t)BS * NA * NC;
    float* out_tgt    = out_fg     + (size_t)BS * NA;

    // Workspace layout.
    const size_t CELLS = (size_t)BS * NB * NA;   // 17,203,200
    char* ws = (char*)d_ws;
    float*         g_align = (float*)ws;                          // CELLS f32
    float*         g_ovl   = (float*)(ws + CELLS * 4);            // CELLS f32
    unsigned char* mtop    = (unsigned char*)(ws + CELLS * 8);    // CELLS u8
    unsigned int*  pos_a   = (unsigned int*)(ws + CELLS * 9);     // BS*NB u32
    unsigned int*  pos_o   = pos_a + BS * NB;                     // BS*NB u32

    k_init<<<dim3((2 * BS * NB + 255) / 256), dim3(256), 0, stream>>>(pos_a, 2 * BS * NB);

    k_align<<<dim3(NA / 16, NB / 16, BS), dim3(32), 0, stream>>>(
        pd_scores, pd_bboxes, anc, gt_labels, gt_bboxes, mask_gt, g_align, g_ovl);

    k_topk<<<dim3(NB, BS), dim3(256), 0, stream>>>(g_align, mask_gt, mtop);

    k_resolve<<<dim3((NA + 255) / 256, BS), dim3(256), 0, stream>>>(
        gt_bboxes, gt_labels, mask_gt, anc, mtop, g_align, g_ovl,
        out_labels, out_bboxes, out_fg, out_tgt, pos_a, pos_o);

    k_scores<<<dim3((NA + 255) / 256, BS), dim3(256), 0, stream>>>(
        g_align, pos_a, pos_o, out_fg, out_tgt, out_labels, out_scores);
}